// Feature_Split_Network_73426760892822
// MI455X (gfx1250) — compile-verified
//
#include <hip/hip_runtime.h>
#include <hip/hip_bf16.h>

#define N_NODES   20000
#define N_EDGES   320000
#define N_GRAPHS  128
#define BN_EPSF   1e-5f

typedef __attribute__((ext_vector_type(2))) float v2f;
typedef __attribute__((ext_vector_type(8))) float v8f;

__device__ __forceinline__ void atomAddF(float* p, float v) {
    unsafeAtomicAdd(p, v);   // hardware global_atomic_add_f32
}

// ---------------------------------------------------------------------------
// y[r, c] = (c < Cin) ? x[r, c] : 0   for c in [0, Cout)  (pad-aware copy)
// ---------------------------------------------------------------------------
__global__ void k_copy_pad(const float* __restrict__ x, float* __restrict__ y,
                           int rows, int Cin, int Cout) {
    long long i = (long long)blockIdx.x * blockDim.x + threadIdx.x;
    if (i >= (long long)rows * Cout) return;
    int r = (int)(i / Cout);
    int c = (int)(i % Cout);
    y[i] = (c < Cin) ? x[(long long)r * Cin + c] : 0.0f;
}

// ---------------------------------------------------------------------------
// y[dst[e], c] += x[src[e], c]   (x stride Cin, y stride Cout)
// ---------------------------------------------------------------------------
__global__ void k_edge_aggr(const float* __restrict__ x, const int* __restrict__ src,
                            const int* __restrict__ dst, float* __restrict__ y,
                            long long total, int Cin, int Cout) {
    long long i = (long long)blockIdx.x * blockDim.x + threadIdx.x;
    if (i >= total) return;
    int e = (int)(i / Cin);
    int c = (int)(i % Cin);
    atomAddF(&y[(long long)dst[e] * Cout + c], x[(long long)src[e] * Cin + c]);
}

// W1_0 pad: out[8,32], rows 6..7 zero
__global__ void k_pad_w1(const float* __restrict__ W, float* __restrict__ out) {
    int i = threadIdx.x;                 // 256 threads == 8*32
    int r = i >> 5;
    out[i] = (r < 6) ? W[i] : 0.0f;
}

// ---------------------------------------------------------------------------
// FP32 WMMA GEMM:  C[M,Nc] = A[M,K] @ B[K,Nc] + bias (+ optional ReLU)
// One wave32 computes a 32x32 macro tile = 2x2 grid of 16x16 WMMA tiles.
// Requires: M % 32 == 0, Nc % 32 == 0, K % 4 == 0 (guaranteed by padding).
// A-frag: lanes 0-15 hold K={k,k+1}, lanes 16-31 hold K={k+2,k+3} (row = lane&15)
// B-frag: same K split, column = lane&15
// C/D:    acc[v] = C[tile_m*16 + (lane>>4)*8 + v][tile_n*16 + (lane&15)]
// ---------------------------------------------------------------------------
template <bool RELU>
__global__ void k_wmma_gemm(const float* __restrict__ A, const float* __restrict__ B,
                            const float* __restrict__ bias, float* __restrict__ C,
                            int M, int K, int Nc) {
    int wave   = (blockIdx.x * blockDim.x + threadIdx.x) >> 5;
    int lane   = threadIdx.x & 31;
    int tilesM = M >> 5;
    int tilesN = Nc >> 5;
    if (wave >= tilesM * tilesN) return;      // wave-uniform: EXEC all-1s at WMMA
    int tm = wave / tilesN;
    int tn = wave - tm * tilesN;

    int half = lane >> 4;                     // 0: K={k,k+1}, 1: K={k+2,k+3}
    int l16  = lane & 15;
    int row0 = tm * 32 + l16;
    int col0 = tn * 32 + l16;
    int col1 = col0 + 16;

    const float* __restrict__ Ap0 = A + (long long)row0 * K + half * 2;
    const float* __restrict__ Ap1 = Ap0 + (long long)16 * K;
    const float* __restrict__ Bp  = B + (long long)(half * 2) * Nc;
    long long bstep = (long long)4 * Nc;

    v8f acc00 = {}, acc01 = {}, acc10 = {}, acc11 = {};
#pragma unroll 2
    for (int k = 0; k < K; k += 4) {
        v2f a0 = *(const v2f*)Ap0;            // global_load_b64 (8B aligned: K even)
        v2f a1 = *(const v2f*)Ap1;
        v2f b0, b1;
        b0.x = Bp[col0];  b0.y = Bp[Nc + col0];
        b1.x = Bp[col1];  b1.y = Bp[Nc + col1];
        acc00 = __builtin_amdgcn_wmma_f32_16x16x4_f32(false, a0, false, b0, (short)0, acc00, false, false);
        acc01 = __builtin_amdgcn_wmma_f32_16x16x4_f32(false, a0, false, b1, (short)0, acc01, false, false);
        acc10 = __builtin_amdgcn_wmma_f32_16x16x4_f32(false, a1, false, b0, (short)0, acc10, false, false);
        acc11 = __builtin_amdgcn_wmma_f32_16x16x4_f32(false, a1, false, b1, (short)0, acc11, false, false);
        Ap0 += 4; Ap1 += 4; Bp += bstep;
    }

    float bv0 = bias[col0];
    float bv1 = bias[col1];
    int ra = tm * 32 + half * 8;
#pragma unroll
    for (int v = 0; v < 8; ++v) {
        float o;
        long long base0 = (long long)(ra + v) * Nc;
        long long base1 = (long long)(ra + 16 + v) * Nc;
        o = acc00[v] + bv0; if (RELU) o = fmaxf(o, 0.0f); C[base0 + col0] = o;
        o = acc01[v] + bv1; if (RELU) o = fmaxf(o, 0.0f); C[base0 + col1] = o;
        o = acc10[v] + bv0; if (RELU) o = fmaxf(o, 0.0f); C[base1 + col0] = o;
        o = acc11[v] + bv1; if (RELU) o = fmaxf(o, 0.0f); C[base1 + col1] = o;
    }
}

// ---------------------------------------------------------------------------
// BatchNorm: per-channel sum / sum-of-squares (128 rows per block, 1 lane/channel)
// ---------------------------------------------------------------------------
__global__ void k_bn_stats(const float* __restrict__ h, float* __restrict__ sum,
                           float* __restrict__ sq, int M, int C) {
    int c  = threadIdx.x;                 // blockDim.x == C (32/64/128/256)
    int r0 = blockIdx.x * 128;
    int r1 = min(r0 + 128, M);
    float s = 0.0f, q = 0.0f;
    for (int r = r0; r < r1; ++r) {
        float v = h[(long long)r * C + c];
        s += v; q += v * v;
    }
    atomAddF(&sum[c], s);
    atomAddF(&sq[c], q);
}

__global__ void k_bn_final(float* __restrict__ stats, const float* __restrict__ gamma,
                           const float* __restrict__ beta, int M, int C) {
    int c = threadIdx.x;
    if (c >= C) return;
    float invM  = 1.0f / (float)M;
    float mu    = stats[c] * invM;
    float var   = stats[512 + c] * invM - mu * mu;
    float scale = gamma[c] * rsqrtf(var + BN_EPSF);
    stats[1024 + c] = scale;
    stats[1536 + c] = beta[c] - mu * scale;
}

__global__ void k_bn_apply(float* __restrict__ h, const float* __restrict__ scale,
                           const float* __restrict__ shift, long long n, int C) {
    long long i = (long long)blockIdx.x * blockDim.x + threadIdx.x;
    if (i >= n) return;
    int c = (int)(i % C);
    h[i] = fmaxf(fmaf(h[i], scale[c], shift[c]), 0.0f);
}

// ---------------------------------------------------------------------------
// Global mean pool over graphs
// ---------------------------------------------------------------------------
__global__ void k_pool_sum(const float* __restrict__ h, const int* __restrict__ batch,
                           float* __restrict__ sums, long long total) {
    long long i = (long long)blockIdx.x * blockDim.x + threadIdx.x;
    if (i >= total) return;
    int n = (int)(i >> 9);
    int c = (int)(i & 511);
    atomAddF(&sums[(long long)batch[n] * 512 + c], h[i]);
}

__global__ void k_pool_cnt(const int* __restrict__ batch, float* __restrict__ cnt, int n) {
    int i = blockIdx.x * blockDim.x + threadIdx.x;
    if (i < n) atomAddF(&cnt[batch[i]], 1.0f);
}

__global__ void k_pool_final(const float* __restrict__ sums, const float* __restrict__ cnt,
                             const float* __restrict__ score, float* __restrict__ cgf_ws,
                             float* __restrict__ out) {
    int i = blockIdx.x * blockDim.x + threadIdx.x;
    if (i >= N_GRAPHS * 512) return;
    int g = i >> 9;
    float gf = sums[i] / fmaxf(cnt[g], 1.0f);
    float sc = score[g];
    float cv = gf * sc;
    float sv = gf * (1.0f - sc);
    out[N_GRAPHS * 43 + i]                  = cv;   // c_graph_feat
    out[N_GRAPHS * 43 + N_GRAPHS * 512 + i] = sv;   // s_graph_feat
    cgf_ws[i] = cv;
}

__global__ void k_head(const float* __restrict__ cgf, const float* __restrict__ W,
                       const float* __restrict__ b, float* __restrict__ out) {
    int i = blockIdx.x * blockDim.x + threadIdx.x;
    if (i >= N_GRAPHS * 43) return;
    int g = i / 43, o = i - g * 43;
    float acc = b[o];
    const float* row = cgf + (long long)g * 512;
    for (int k = 0; k < 512; ++k) acc = fmaf(row[k], W[(long long)k * 43 + o], acc);
    out[i] = acc;
}

// ---------------------------------------------------------------------------
extern "C" void kernel_launch(void* const* d_in, const int* in_sizes, int n_in,
                              void* d_out, int out_size, void* d_ws, size_t ws_size,
                              hipStream_t stream) {
    const float* graph_x = (const float*)d_in[0];
    const int*   eidx    = (const int*)d_in[1];     // (2, E): [0:E]=src, [E:2E]=dst
    const int*   batch   = (const int*)d_in[2];
    const float* score   = (const float*)d_in[3];

    const int din [4] = {6,   64,  128, 256};   // true input widths
    const int dpad[4] = {8,   64,  128, 256};   // K fed to GEMM1 (layer 0 padded)
    const int dh  [4] = {32,  64,  128, 256};
    const int dout[4] = {64, 128,  256, 512};

    const float *W1[4], *b1[4], *gm[4], *bt[4], *W2[4], *b2[4];
    for (int l = 0; l < 4; ++l) {
        int base = 4 + 6 * l;
        W1[l] = (const float*)d_in[base + 0];
        b1[l] = (const float*)d_in[base + 1];
        gm[l] = (const float*)d_in[base + 2];
        bt[l] = (const float*)d_in[base + 3];
        W2[l] = (const float*)d_in[base + 4];
        b2[l] = (const float*)d_in[base + 5];
    }
    const float* W_out = (const float*)d_in[28];
    const float* b_out = (const float*)d_in[29];

    const int* src = eidx;
    const int* dst = eidx + N_EDGES;

    // workspace layout
    float* bufA  = (float*)d_ws;                          // N x 512 (ping)
    float* bufB  = bufA  + (size_t)N_NODES * 512;         // N x 512 (pong)
    float* bufH  = bufB  + (size_t)N_NODES * 512;         // N x 256 (hidden / c_gf stage)
    float* stats = bufH  + (size_t)N_NODES * 256;         // 4 x 512 (sum|sq|scale|shift)
    float* psum  = stats + 4 * 512;                       // G x 512
    float* pcnt  = psum  + (size_t)N_GRAPHS * 512;        // G
    float* w1pad = pcnt  + N_GRAPHS;                      // 8 x 32 padded W1_0

    auto gemm = [&](const float* A, const float* B, const float* bias, float* C,
                    int M, int K, int Nc, bool relu) {
        int tiles  = (M >> 5) * (Nc >> 5);                // 32x32 macro tiles
        int blocks = (tiles + 7) / 8;                     // 8 waves (wave32) per block
        if (relu) k_wmma_gemm<true ><<<blocks, 256, 0, stream>>>(A, B, bias, C, M, K, Nc);
        else      k_wmma_gemm<false><<<blocks, 256, 0, stream>>>(A, B, bias, C, M, K, Nc);
    };

    // one-off: zero-pad W1_0 (6x32) -> w1pad (8x32)
    k_pad_w1<<<1, 256, 0, stream>>>(W1[0], w1pad);

    const float* x = graph_x;
    for (int l = 0; l < 4; ++l) {
        float* scratch = (l & 1) ? bufB : bufA;           // y, later reused as layer output

        // y = x + sum_{edges} x[src]   (layer 0 written with stride 8, cols 6..7 = 0)
        long long nEl = (long long)N_NODES * dpad[l];
        k_copy_pad<<<(unsigned)((nEl + 255) / 256), 256, 0, stream>>>(x, scratch, N_NODES, din[l], dpad[l]);
        long long tot = (long long)N_EDGES * din[l];
        k_edge_aggr<<<(unsigned)((tot + 255) / 256), 256, 0, stream>>>(x, src, dst, scratch, tot, din[l], dpad[l]);

        // h = y @ W1 + b1
        const float* Wg1 = (l == 0) ? w1pad : W1[l];
        gemm(scratch, Wg1, b1[l], bufH, N_NODES, dpad[l], dh[l], false);

        // BatchNorm (training mode, biased var) + ReLU
        hipMemsetAsync(stats, 0, 1024 * sizeof(float), stream);
        k_bn_stats<<<(N_NODES + 127) / 128, dh[l], 0, stream>>>(bufH, stats, stats + 512, N_NODES, dh[l]);
        k_bn_final<<<1, dh[l], 0, stream>>>(stats, gm[l], bt[l], N_NODES, dh[l]);
        long long nH = (long long)N_NODES * dh[l];
        k_bn_apply<<<(unsigned)((nH + 255) / 256), 256, 0, stream>>>(bufH, stats + 1024, stats + 1536, nH, dh[l]);

        // out = relu(h @ W2 + b2)  (overwrites y — y is dead by now)
        gemm(bufH, W2[l], b2[l], scratch, N_NODES, dh[l], dout[l], true);
        x = scratch;
    }

    // global mean pool + score split + head
    hipMemsetAsync(psum, 0, (size_t)(N_GRAPHS * 512 + N_GRAPHS) * sizeof(float), stream);
    long long npool = (long long)N_NODES * 512;
    k_pool_sum<<<(unsigned)((npool + 255) / 256), 256, 0, stream>>>(x, batch, psum, npool);
    k_pool_cnt<<<(N_NODES + 255) / 256, 256, 0, stream>>>(batch, pcnt, N_NODES);
    k_pool_final<<<(N_GRAPHS * 512 + 255) / 256, 256, 0, stream>>>(psum, pcnt, score, bufH, (float*)d_out);
    k_head<<<(N_GRAPHS * 43 + 255) / 256, 256, 0, stream>>>(bufH, W_out, b_out, (float*)d_out);
}